// WeightedHausdorffDistance_59511066853607
// MI455X (gfx1250) — compile-verified
//
#include <hip/hip_runtime.h>

// Weighted Hausdorff distance on MI455X (gfx1250, wave32).
// d^2(i,j) computed as a rank-4 GEMM via V_WMMA_F32_16X16X4_F32:
//   A(16x4)=(y,x,1,0), B(4x16)=(-2gy,-2gx,gy^2+gx^2,0), C=row term y^2+x^2
// -> D = exact squared distances (integers < 2^20, exact in f32).
// GT-derived B-matrix values are staged once per block into LDS in lane
// layout, so the 32-tile inner loop is ds_load_b64 -> wmma -> min tree ->
// ds_min_u32 with no VMEM waits and no cross-lane shuffles.
// Term1: min_j d = sqrt(min_j d^2), one sqrt/pixel.
// Term2: min_i (d+eps)*w_i approximated by sqrt(min_i d^2*w_i^2), one sqrt/point.

typedef __attribute__((ext_vector_type(2))) float v2f;
typedef __attribute__((ext_vector_type(8))) float v8f;

#define WHD_EPS   1e-6f
#define WHD_MAXD  362.03866599f            /* sqrt(256^2+256^2) */
#define WHD_CDEN  (WHD_EPS / WHD_MAXD)     /* eps / max_dist */

#define NB 4
#define HH 256
#define WW 256
#define NPIX (HH*WW)
#define MM 512

__global__ __launch_bounds__(256) void whd_init(unsigned* __restrict__ minq,
                                                float* __restrict__ sums) {
  int t = blockIdx.x * 256 + threadIdx.x;
  if (t < NB * MM) minq[t] = 0x7F800000u;   // +inf bit pattern (uint-ordered for >=0 floats)
  if (t < NB * 2)  sums[t] = 0.0f;
}

__global__ __launch_bounds__(256) void whd_main(const float* __restrict__ pm,
                                                const int*   __restrict__ gt,
                                                unsigned*    __restrict__ gminq,
                                                float*       __restrict__ gsums) {
  const int b    = blockIdx.y;
  const int lane = threadIdx.x & 31;
  const int wv   = threadIdx.x >> 5;                 // wave in block: 0..7
  const int strip = blockIdx.x * 8 + wv;             // 16-pixel strip: 0..4095
  const int base  = strip * 16;                      // flat pixel base
  const int y     = base >> 8;                       // image row (W=256)
  const int x0    = base & 255;                      // starting column

  __shared__ unsigned lminq[MM];                     // block-level term2 mins
  __shared__ float2   blo[MM];                       // (-2gy, -2gx) per GT point
  __shared__ float2   bhi[MM];                       // (gy^2+gx^2, 0) per GT point
  __shared__ float    w2s[8][16];                    // per-wave w_i^2
  __shared__ float    ps [8][16];                    // per-wave p_i

  // Stage GT points once per block, pre-transformed into B-matrix lane layout.
  for (int j = threadIdx.x; j < MM; j += 256) {
    lminq[j] = 0x7F800000u;
    int2 g = ((const int2*)gt)[b * MM + j];          // (row, col)
    float gy = (float)g.x, gx = (float)g.y;
    blo[j] = make_float2(-2.0f * gy, -2.0f * gx);
    bhi[j] = make_float2(gy * gy + gx * gx, 0.0f);
  }

  float p_l = 0.0f;
  if (lane < 16) {
    p_l = pm[b * NPIX + base + lane];
    float p2 = p_l * p_l;
    float p4 = p2 * p2;
    float w  = 1.0f / (p4 + WHD_CDEN);
    ps [wv][lane] = p_l;
    w2s[wv][lane] = w * w;
  }
  __syncthreads();

  const int hi = lane >> 4;                          // lane half selects K/M halves
  const int jl = lane & 15;

  // A matrix (16x4 f32): VGPR0 = K0 (lanes 0-15) / K2 (lanes 16-31), VGPR1 = K1/K3
  v2f a;
  a.x = hi ? 1.0f : (float)y;                        // K0 = y   | K2 = 1
  a.y = hi ? 0.0f : (float)(x0 + lane);              // K1 = x   | K3 = 0

  // C matrix preload: C[i][j] = y^2 + x_i^2, i = v + 8*hi (pure per-lane arithmetic)
  v8f c;
  const float yf2 = (float)(y * y);
#pragma unroll
  for (int v = 0; v < 8; ++v) {
    float xi = (float)(x0 + v + 8 * hi);
    c[v] = yf2 + xi * xi;
  }

  float w2v[8];
#pragma unroll
  for (int v = 0; v < 8; ++v) w2v[v] = w2s[wv][v + 8 * hi];

  float md2[8];
#pragma unroll
  for (int v = 0; v < 8; ++v) md2[v] = 3.0e38f;

  // Per-lane B source: low-half lanes read blo[], high-half lanes read bhi[].
  const float2* bsrc = hi ? bhi : blo;

#pragma unroll 4
  for (int t = 0; t < 32; ++t) {                     // 32 j-tiles of 16 GT points
    const int jj = t * 16 + jl;
    float2 bv = bsrc[jj];                            // ds_load_b64, linear offset
    v2f bb;
    bb.x = bv.x;
    bb.y = bv.y;

    // D = A*B + C = exact squared distances for this 16x16 tile
    v8f d = __builtin_amdgcn_wmma_f32_16x16x4_f32(
        false, a, false, bb, (short)0, c, false, false);

    float q = 3.0e38f;
#pragma unroll
    for (int v = 0; v < 8; ++v) {
      float dv = d[v];
      md2[v] = fminf(md2[v], dv);                    // term1: min over j per pixel
      q = fminf(q, dv * w2v[v]);                     // term2: weighted min over i
    }
    // All 32 lanes atomic-min (two lanes per j) -- LDS combines the i-halves.
    atomicMin(&lminq[jj], __float_as_uint(q));
  }

  // term1: cross-lane min over the 16 j-lanes for each of this wave's 16 pixels
  float acc = 0.0f;
#pragma unroll
  for (int v = 0; v < 8; ++v) {
    float m = md2[v];
    m = fminf(m, __shfl_xor(m, 1, 32));
    m = fminf(m, __shfl_xor(m, 2, 32));
    m = fminf(m, __shfl_xor(m, 4, 32));
    m = fminf(m, __shfl_xor(m, 8, 32));
    if (jl == v) acc = ps[wv][v + 8 * hi] * sqrtf(m);   // one pixel per (lane<8, half)
  }
  float sp = (lane < 16) ? p_l : 0.0f;
#pragma unroll
  for (int mask = 1; mask < 32; mask <<= 1) {
    acc += __shfl_xor(acc, mask, 32);
    sp  += __shfl_xor(sp,  mask, 32);
  }
  if (lane == 0) {
    atomicAdd(&gsums[b * 2 + 0], sp);
    atomicAdd(&gsums[b * 2 + 1], acc);
  }

  __syncthreads();
  for (int t = threadIdx.x; t < MM; t += 256)
    atomicMin(&gminq[b * MM + t], lminq[t]);
}

__global__ __launch_bounds__(256) void whd_final(const unsigned* __restrict__ gminq,
                                                 const float*    __restrict__ gsums,
                                                 float*          __restrict__ out) {
  __shared__ float red[256];
  const int tid = threadIdx.x;
  float t2 = 0.0f;
  for (int b = 0; b < NB; ++b) {
    float local = 0.0f;
    for (int j = tid; j < MM; j += 256) {
      float q = __uint_as_float(gminq[b * MM + j]);
      float dmin = fminf(sqrtf(q), WHD_MAXD);        // clamp to [0, max_dist]
      local += dmin;
    }
    red[tid] = local;
    __syncthreads();
    for (int s = 128; s > 0; s >>= 1) {
      if (tid < s) red[tid] += red[tid + s];
      __syncthreads();
    }
    if (tid == 0) t2 += red[0] / (float)MM;
    __syncthreads();
  }
  if (tid == 0) {
    float t1 = 0.0f;
    for (int b = 0; b < NB; ++b)
      t1 += gsums[b * 2 + 1] / (gsums[b * 2 + 0] + WHD_EPS);
    out[0] = t1 / (float)NB + t2 / (float)NB;
  }
}

extern "C" void kernel_launch(void* const* d_in, const int* in_sizes, int n_in,
                              void* d_out, int out_size, void* d_ws, size_t ws_size,
                              hipStream_t stream) {
  const float* pm = (const float*)d_in[0];           // prob_map (4,256,256) f32
  const int*   gt = (const int*)d_in[1];             // gt (4,512,2) i32
  unsigned* minq = (unsigned*)d_ws;                  // [4*512] term2 mins (uint bits)
  float*    sums = (float*)((char*)d_ws + NB * MM * sizeof(unsigned)); // [4*2]
  float*    out  = (float*)d_out;

  whd_init <<<8, 256, 0, stream>>>(minq, sums);
  whd_main <<<dim3(512, NB), 256, 0, stream>>>(pm, gt, minq, sums);
  whd_final<<<1, 256, 0, stream>>>(minq, sums, out);
}